// MultiViewLoss_64665027608843
// MI455X (gfx1250) — compile-verified
//
#include <hip/hip_runtime.h>
#include <cmath>

// ---------------- problem constants (fixed by the reference) ----------------
constexpr int   Nn          = 2048;   // batch size
constexpr int   Dd          = 1024;   // feature dim
constexpr float GAMMA_C     = 1000.0f;
constexpr float WLAMBDA_C   = 0.1f;
constexpr float EPS_C       = 0.05f;
constexpr int   GW_OUTER_C  = 5;
constexpr int   SINK_IT_C   = 30;

constexpr int BM = 64, BN = 64, BK = 32;   // GEMM macro tile
constexpr int TPB = 256;                   // 8 wave32s per block

typedef float v2f __attribute__((ext_vector_type(2)));
typedef float v8f __attribute__((ext_vector_type(8)));

enum { EPI_RBF = 0, EPI_MCOST = 1, EPI_STORE = 2, EPI_BARY = 3 };

// ---------------------------------------------------------------------------
// Tiled f32 WMMA GEMM:  C = op(A) * op(B)  with fused epilogues.
//   TRANS_A: logical A[m,k] = Asrc[k*lda + m]   (used for T^T)
//   TRANS_B: logical B[k,n] = Bsrc[n*ldb + k]   (used for X^T in Gram)
// Each block computes a 64x64 tile; each wave owns a 16x32 strip (2 WMMA
// accumulators, A fragment reused across the B pair).
// ---------------------------------------------------------------------------
template <bool TRANS_A, bool TRANS_B, int EPI>
__global__ __launch_bounds__(TPB)
void wmma_gemm(const float* __restrict__ A, int lda,
               const float* __restrict__ B, int ldb,
               float* __restrict__ Cout, int ldc, int Kdim,
               const float* __restrict__ rowvec,   // sq[] (RBF) or a[] (MCOST)
               const float* __restrict__ colvec,   // sq[] (RBF)
               const float* __restrict__ wbuf, int widx, int accum)
{
    __shared__ float As[BM][BK + 1];   // +1 pad: conflict-free transposed staging
    __shared__ float Bs[BK][BN + 1];

    const int tid  = threadIdx.x;
    const int row0 = blockIdx.y * BM;
    const int col0 = blockIdx.x * BN;

    const int wave  = tid >> 5;
    const int lane  = tid & 31;
    const int lr    = lane & 15;          // m (A frag) / n (B frag, C cols)
    const int khalf = (lane >> 4) << 1;   // K sub-offset: 0 or 2
    const int jrow  = (lane >> 4) << 3;   // C/D row offset: 0 or 8

    const int tm  = wave & 3;             // tile row within block: 0..3
    const int tn0 = (wave >> 2) << 1;     // tile col pair base: 0 or 2
    const int tn1 = tn0 + 1;

    v8f acc0 = {};
    v8f acc1 = {};

    for (int kb = 0; kb < Kdim; kb += BK) {
        // ---- stage A panel: As[m][k] = Aelem(row0+m, kb+k) ----
        for (int idx = tid; idx < BM * BK; idx += TPB) {
            if (!TRANS_A) {
                int m = idx / BK, k = idx % BK;
                As[m][k] = A[(row0 + m) * lda + (kb + k)];
            } else {
                int k = idx / BM, m = idx % BM;   // coalesced over m
                As[m][k] = A[(kb + k) * lda + (row0 + m)];
            }
        }
        // ---- stage B panel: Bs[k][n] = Belem(kb+k, col0+n) ----
        for (int idx = tid; idx < BK * BN; idx += TPB) {
            if (!TRANS_B) {
                int k = idx / BN, n = idx % BN;
                Bs[k][n] = B[(kb + k) * ldb + (col0 + n)];
            } else {
                int n = idx / BK, k = idx % BK;   // coalesced over k
                Bs[k][n] = B[(col0 + n) * ldb + (kb + k)];
            }
        }
        __syncthreads();

        #pragma unroll
        for (int kk = 0; kk < BK; kk += 4) {
            // A 16x4 fragment: lane holds {A[m,k0], A[m,k0+1]}, m=lane%16,
            // k0 = kk + (lane/16)*2  (ISA 32-bit A layout)
            v2f av, b0, b1;
            av[0] = As[tm * 16 + lr][kk + khalf];
            av[1] = As[tm * 16 + lr][kk + khalf + 1];
            // B 4x16 fragment: lane holds {B[k0,n], B[k0+1,n]}, n=lane%16
            b0[0] = Bs[kk + khalf][tn0 * 16 + lr];
            b0[1] = Bs[kk + khalf + 1][tn0 * 16 + lr];
            b1[0] = Bs[kk + khalf][tn1 * 16 + lr];
            b1[1] = Bs[kk + khalf + 1][tn1 * 16 + lr];
            acc0 = __builtin_amdgcn_wmma_f32_16x16x4_f32(
                       false, av, false, b0, (short)0, acc0, false, false);
            acc1 = __builtin_amdgcn_wmma_f32_16x16x4_f32(
                       false, av, false, b1, (short)0, acc1, false, false);
        }
        __syncthreads();
    }

    // ---- fused epilogue (C/D layout: VGPR j -> row j + jrow, col = lane%16) ----
    const float invn   = 1.0f / (float)Nn;
    const float inveps = 1.0f / EPS_C;
    const float grbf   = GAMMA_C / (float)Dd;
    float wv = 0.0f;
    if (EPI == EPI_BARY) wv = wbuf[widx];

    #pragma unroll
    for (int j = 0; j < 8; ++j) {
        const int gr = row0 + tm * 16 + jrow + j;
        #pragma unroll
        for (int t = 0; t < 2; ++t) {
            const int   tn  = t ? tn1 : tn0;
            const float acc = t ? acc1[j] : acc0[j];
            const int   gc  = col0 + tn * 16 + lr;
            float* dst = Cout + gr * ldc + gc;
            if (EPI == EPI_RBF) {
                float d2 = fmaxf(rowvec[gr] + colvec[gc] - 2.0f * acc, 0.0f);
                *dst = __expf(-grbf * d2);
            } else if (EPI == EPI_MCOST) {
                // M = -(constC - 2S)/eps,  constC[r,c] = a[r] + 1/n  (C2 == I)
                *dst = (2.0f * acc - rowvec[gr] - invn) * inveps;
            } else if (EPI == EPI_STORE) {
                *dst = acc;
            } else {  // EPI_BARY:  Bm (+)= w * (T^T U)
                if (accum) *dst += wv * acc; else *dst = wv * acc;
            }
        }
    }
}

// ---------------- helper kernels ----------------
__global__ void fill_kernel(float* __restrict__ p, float v, int count) {
    int i = blockIdx.x * blockDim.x + threadIdx.x;
    if (i < count) p[i] = v;
}

// out[r] = scale * sum_k row[k]^2   (width = Dd for sq, Nn for a)
__global__ __launch_bounds__(TPB)
void rowsumsq_kernel(const float* __restrict__ X, int width, float scale,
                     float* __restrict__ out) {
    const int r = blockIdx.x;
    const float* row = X + (size_t)r * width;
    float s = 0.0f;
    for (int c = threadIdx.x; c < width; c += TPB) { float x = row[c]; s += x * x; }
    __shared__ float sh[TPB];
    sh[threadIdx.x] = s; __syncthreads();
    for (int st = TPB / 2; st > 0; st >>= 1) {
        if (threadIdx.x < st) sh[threadIdx.x] += sh[threadIdx.x + st];
        __syncthreads();
    }
    if (threadIdx.x == 0) out[r] = sh[0] * scale;
}

// f[r] = logp - logsumexp_c( M[r,c] + g[c] )   (one block per row, online LSE)
__global__ __launch_bounds__(TPB)
void row_lse_kernel(const float* __restrict__ Mmat, const float* __restrict__ g,
                    float* __restrict__ f, float logp) {
    const int r = blockIdx.x;
    const float* row = Mmat + (size_t)r * Nn;
    float mx = -INFINITY, s = 0.0f;
    for (int c = threadIdx.x; c < Nn; c += TPB) {
        float x = row[c] + g[c];
        if (x > mx) { s *= __expf(mx - x); mx = x; }
        s += __expf(x - mx);
    }
    __shared__ float smx[TPB], ss[TPB];
    smx[threadIdx.x] = mx; ss[threadIdx.x] = s; __syncthreads();
    for (int st = TPB / 2; st > 0; st >>= 1) {
        if (threadIdx.x < st) {
            float m1 = smx[threadIdx.x], m2 = smx[threadIdx.x + st];
            float s1 = ss[threadIdx.x],  s2 = ss[threadIdx.x + st];
            float mm = fmaxf(m1, m2);
            ss[threadIdx.x]  = s1 * __expf(m1 - mm) + s2 * __expf(m2 - mm);
            smx[threadIdx.x] = mm;
        }
        __syncthreads();
    }
    if (threadIdx.x == 0) f[r] = logp - (smx[0] + __logf(ss[0]));
}

// g[c] = logq - logsumexp_r( M[r,c] + f[r] )   (one thread per column, coalesced)
__global__ __launch_bounds__(TPB)
void col_lse_kernel(const float* __restrict__ Mmat, const float* __restrict__ f,
                    float* __restrict__ g, float logq) {
    const int c = blockIdx.x * TPB + threadIdx.x;
    float mx = -INFINITY, s = 0.0f;
    for (int r = 0; r < Nn; ++r) {
        float x = Mmat[(size_t)r * Nn + c] + f[r];
        if (x > mx) { s *= __expf(mx - x); mx = x; }
        s += __expf(x - mx);
    }
    g[c] = logq - (mx + __logf(s));
}

// T = exp(M + f[:,None] + g[None,:])
__global__ __launch_bounds__(TPB)
void t_update_kernel(const float* __restrict__ Mmat, const float* __restrict__ f,
                     const float* __restrict__ g, float* __restrict__ T) {
    size_t i = (size_t)blockIdx.x * TPB + threadIdx.x;
    if (i >= (size_t)Nn * Nn) return;
    int r = (int)(i >> 11);       // /2048
    int c = (int)(i & (Nn - 1));
    T[i] = __expf(Mmat[i] + f[r] + g[c]);
}

// w = softmax(weights); wbuf = {w0, w1, sum w*log(w+1e-10)}
__global__ void softmax_w_kernel(const float* __restrict__ wts, float* __restrict__ wbuf) {
    if (threadIdx.x == 0 && blockIdx.x == 0) {
        float a0 = wts[0], a1 = wts[1];
        float mx = fmaxf(a0, a1);
        float e0 = __expf(a0 - mx), e1 = __expf(a1 - mx);
        float inv = 1.0f / (e0 + e1);
        float w0 = e0 * inv, w1 = e1 * inv;
        wbuf[0] = w0; wbuf[1] = w1;
        wbuf[2] = w0 * __logf(w0 + 1e-10f) + w1 * __logf(w1 + 1e-10f);
    }
}

// partial[r] = sum_c (Bm[r,c]*n^2 - (r==c))^2     (deterministic 2-stage mean)
__global__ __launch_bounds__(TPB)
void loss_row_kernel(const float* __restrict__ Bm, float* __restrict__ partial) {
    const int r = blockIdx.x;
    const float nsq = (float)Nn * (float)Nn;
    float s = 0.0f;
    for (int c = threadIdx.x; c < Nn; c += TPB) {
        float v = Bm[(size_t)r * Nn + c] * nsq - (c == r ? 1.0f : 0.0f);
        s += v * v;
    }
    __shared__ float sh[TPB];
    sh[threadIdx.x] = s; __syncthreads();
    for (int st = TPB / 2; st > 0; st >>= 1) {
        if (threadIdx.x < st) sh[threadIdx.x] += sh[threadIdx.x + st];
        __syncthreads();
    }
    if (threadIdx.x == 0) partial[r] = sh[0];
}

__global__ __launch_bounds__(TPB)
void loss_final_kernel(const float* __restrict__ partial, const float* __restrict__ wbuf,
                       float* __restrict__ out) {
    double s = 0.0;
    for (int i = threadIdx.x; i < Nn; i += TPB) s += (double)partial[i];
    __shared__ double sh[TPB];
    sh[threadIdx.x] = s; __syncthreads();
    for (int st = TPB / 2; st > 0; st >>= 1) {
        if (threadIdx.x < st) sh[threadIdx.x] += sh[threadIdx.x + st];
        __syncthreads();
    }
    if (threadIdx.x == 0)
        out[0] = (float)(sh[0] / ((double)Nn * (double)Nn)) + WLAMBDA_C * wbuf[2];
}

// ---------------------------------------------------------------------------
extern "C" void kernel_launch(void* const* d_in, const int* in_sizes, int n_in,
                              void* d_out, int out_size, void* d_ws, size_t ws_size,
                              hipStream_t stream) {
    const float* tx  = (const float*)d_in[0];
    const float* ph  = (const float*)d_in[1];
    const float* wts = (const float*)d_in[2];

    const size_t nn = (size_t)Nn * Nn;
    float* ws  = (float*)d_ws;
    float* Cv  = ws;            // [n,n] RBF kernel of current view
    float* T   = ws + nn;       // [n,n] transport plan
    float* Mm  = ws + 2 * nn;   // [n,n] Sinkhorn log-cost M; reused as U = Cv*T
    float* Bm  = ws + 3 * nn;   // [n,n] barycenter accumulator
    float* sq  = ws + 4 * nn;   // [n]   row squared norms
    float* av  = sq + Nn;       // [n]   a = (Cv^2) @ p
    float* fv  = av + Nn;       // [n]   Sinkhorn f   (fv,gv adjacent -> one fill)
    float* gv  = fv + Nn;       // [n]   Sinkhorn g
    float* wb  = gv + Nn;       // [3]   softmax weights + entropy
    float* prt = wb + 16;       // [n]   loss row partials
    // total workspace: 4*nn + ~5n floats  ~= 67.2 MB

    const float invn  = 1.0f / (float)Nn;
    const float logp  = -logf((float)Nn);
    const dim3  ggrid(Nn / BN, Nn / BM);
    const int   nblk_nn = (int)((nn + TPB - 1) / TPB);

    softmax_w_kernel<<<1, 32, 0, stream>>>(wts, wb);

    for (int v = 0; v < 2; ++v) {
        const float* X = v ? ph : tx;

        // Cs[v] = exp(-(gamma/D) * cdist^2) via Gram GEMM (X X^T) + fused epilogue
        rowsumsq_kernel<<<Nn, TPB, 0, stream>>>(X, Dd, 1.0f, sq);
        wmma_gemm<false, true, EPI_RBF><<<ggrid, TPB, 0, stream>>>(
            X, Dd, X, Dd, Cv, Nn, Dd, sq, sq, nullptr, 0, 0);

        // a = (Cv .^2) @ p  ;  T0 = p q^T = 1/n^2
        rowsumsq_kernel<<<Nn, TPB, 0, stream>>>(Cv, Nn, invn, av);
        fill_kernel<<<nblk_nn, TPB, 0, stream>>>(T, invn * invn, (int)nn);

        // entropic GW (C2 == I):  M = (2*Cv*T - a[r] - 1/n)/eps, then log-Sinkhorn
        for (int outer = 0; outer < GW_OUTER_C; ++outer) {
            wmma_gemm<false, false, EPI_MCOST><<<ggrid, TPB, 0, stream>>>(
                Cv, Nn, T, Nn, Mm, Nn, Nn, av, nullptr, nullptr, 0, 0);
            fill_kernel<<<(2 * Nn + TPB - 1) / TPB, TPB, 0, stream>>>(fv, 0.0f, 2 * Nn);
            for (int it = 0; it < SINK_IT_C; ++it) {
                row_lse_kernel<<<Nn, TPB, 0, stream>>>(Mm, gv, fv, logp);
                col_lse_kernel<<<Nn / TPB, TPB, 0, stream>>>(Mm, fv, gv, logp);
            }
            t_update_kernel<<<nblk_nn, TPB, 0, stream>>>(Mm, fv, gv, T);
        }

        // Bm (+)= w[v] * T^T (Cv T)   (U stored in Mm, which is dead now)
        wmma_gemm<false, false, EPI_STORE><<<ggrid, TPB, 0, stream>>>(
            Cv, Nn, T, Nn, Mm, Nn, Nn, nullptr, nullptr, nullptr, 0, 0);
        wmma_gemm<true, false, EPI_BARY><<<ggrid, TPB, 0, stream>>>(
            T, Nn, Mm, Nn, Bm, Nn, Nn, nullptr, nullptr, wb, v, v);
    }

    // loss = mean((Bm*n^2 - I)^2) + lambda * entropy
    loss_row_kernel<<<Nn, TPB, 0, stream>>>(Bm, prt);
    loss_final_kernel<<<1, TPB, 0, stream>>>(prt, wb, (float*)d_out);

    (void)in_sizes; (void)n_in; (void)out_size; (void)ws_size;
}